// FluenceVolumeLayer_19524921327963
// MI455X (gfx1250) — compile-verified
//
#include <hip/hip_runtime.h>
#include <hip/hip_bf16.h>
#include <stdint.h>

// TDM descriptor word groups (amdgpu-toolchain / clang-23 6-arg builtin form)
typedef __attribute__((ext_vector_type(4))) unsigned int v4u;
typedef __attribute__((ext_vector_type(8))) int          v8i;
typedef __attribute__((ext_vector_type(4))) int          v4i;

#define BN  8      // batch
#define DN  96     // depths
#define HN  192    // output H
#define WN  192    // output W (== blockDim.x)
#define HT  16     // h-rows per block (double-buffered TDM pipeline)
#define HFD 400    // fluence H
#define WFD 400    // fluence W

// Bilinear tap window + zero-padding-folded weights for one normalized coord.
// s = clamped 2-tap window start; wa/wb = weights for taps at s, s+1.
__device__ __forceinline__ void coefs(float v, int& s, float& wa, float& wb) {
  float ic = ((v + 1.0f) * (float)WFD - 1.0f) * 0.5f;   // same arithmetic as reference
  float fc = floorf(ic);
  float w1 = ic - fc, w0 = 1.0f - w1;
  int   i0 = (int)fc;
  s  = i0 < 0 ? 0 : (i0 > WFD - 2 ? WFD - 2 : i0);
  wa = 0.0f; wb = 0.0f;
  if (i0     >= 0 && i0     < WFD) { if (i0     == s) wa += w0; else wb += w0; }
  if (i0 + 1 >= 0 && i0 + 1 < WFD) { if (i0 + 1 == s) wa += w1; else wb += w1; }
}

// Issue one TDM 2-D descriptor: copy 2 contiguous fluence rows (2x400 f32) to LDS.
__device__ __forceinline__ void tdm_load_2rows(const float* src, unsigned lds_off) {
  unsigned long long ga = (unsigned long long)(uintptr_t)src;
  // D# group 0: count=1 | lds_addr | global_addr[56:0] | type=2
  v4u g0 = { 1u, lds_off, (unsigned)ga,
             (((unsigned)(ga >> 32)) & 0x01FFFFFFu) | 0x80000000u };
  // D# group 1: data_size=4B; tensor 400x2 (== tile); tile 400x2; dim0 stride 400
  v8i g1 = { 0x00020000,                   // data_size=2 (4B), no mask/pad/iterate
             (int)((unsigned)WFD << 16),   // tensor_dim0 lo16 @bits[63:48]
             (int)(2u << 16),              // tensor_dim1 lo16 @bits[95:80]
             (int)((unsigned)WFD << 16),   // tile_dim0 @bits[127:112]
             2,                            // tile_dim1 @bits[143:128]
             WFD,                          // tensor_dim0_stride[31:0]
             0, 0 };
  v4i z4 = { 0, 0, 0, 0 };
  v8i z8 = { 0, 0, 0, 0, 0, 0, 0, 0 };
  __builtin_amdgcn_tensor_load_to_lds(g0, g1, z4, z4, z8, 0);
}

__global__ __launch_bounds__(WN) void FluenceVolumeLayer_19524921327963_kernel(
    const float* __restrict__ fl,    // [B, 400, 400]
    const float* __restrict__ grid,  // [D, W, H, 2]  (separable by construction)
    const float* __restrict__ pc,    // [D]
    float* __restrict__ out)         // [B, D, H, W]
{
  const int h0 = blockIdx.x * HT, d = blockIdx.y, b = blockIdx.z;
  const int w  = threadIdx.x;
  const bool lead = (threadIdx.x < 32);          // wave 0 (wave32) issues TDM

  __shared__ float rows[2][2 * WFD];             // double-buffered row pairs (6.4 KB)
  const unsigned lds0 = (unsigned)(unsigned long long)
      (__attribute__((address_space(3))) float*)&rows[0][0];
  const float* fb = fl + (size_t)b * HFD * WFD;

  // ---- x coefficients: depend on (d,w) only -> computed ONCE per block ----
  int cs; float wxa, wxb;
  coefs(grid[(size_t)(d * WN + w) * HN * 2], cs, wxa, wxb);   // grid[d,w,0,0]
  const float corr = pc[d];

  // ---- prologue: DMA row pair for first h ---------------------------------
  if (lead) {
    int rs; float ta, tb;
    coefs(grid[(d * WN * HN + h0) * 2 + 1], rs, ta, tb);      // grid[d,0,h0,1]
    tdm_load_2rows(fb + (size_t)rs * WFD, lds0);
  }

  // ---- pipelined loop over h-rows -----------------------------------------
  for (int ht = 0; ht < HT; ++ht) {
    const int h = h0 + ht;

    if (lead) {
      if (ht + 1 < HT) {                 // prefetch next row pair into other buffer
        int rsn; float ta, tb;
        coefs(grid[(d * WN * HN + h + 1) * 2 + 1], rsn, ta, tb);
        tdm_load_2rows(fb + (size_t)rsn * WFD,
                       lds0 + (unsigned)(((ht + 1) & 1) * 2 * WFD * 4));
        __builtin_amdgcn_s_wait_tensorcnt(1);   // TDM[ht] done, TDM[ht+1] in flight
      } else {
        __builtin_amdgcn_s_wait_tensorcnt(0);   // last iteration: drain
      }
    }
    __syncthreads();                      // publish TDM[ht] data to all waves

    // vertical (y) coefficients: block-uniform -> compiler scalarizes to SALU
    int rs; float wA, wB;
    coefs(grid[(d * WN * HN + h) * 2 + 1], rs, wA, wB);
    (void)rs;

    const float* rbuf = rows[ht & 1];
    float a0 = rbuf[cs],       a1 = rbuf[cs + 1];        // row rs
    float b0 = rbuf[WFD + cs], b1 = rbuf[WFD + cs + 1];  // row rs+1
    float gA = wA * a0 + wB * b0;                        // vertical lerp @ cs
    float gB = wA * a1 + wB * b1;                        // vertical lerp @ cs+1
    out[(((size_t)b * DN + d) * HN + h) * WN + w] = (wxa * gA + wxb * gB) * corr;

    __syncthreads();                      // compute[ht] done before TDM[ht+2] reuses buffer
  }
}

extern "C" void kernel_launch(void* const* d_in, const int* in_sizes, int n_in,
                              void* d_out, int out_size, void* d_ws, size_t ws_size,
                              hipStream_t stream) {
  (void)in_sizes; (void)n_in; (void)out_size; (void)d_ws; (void)ws_size;
  const float* fl   = (const float*)d_in[0];   // fluence_map   [8,400,400]
  const float* grid = (const float*)d_in[1];   // sampling_grids[96,192,192,2]
  const float* pc   = (const float*)d_in[2];   // profile_corrections [96]
  float* out = (float*)d_out;                  // [8,96,192,192,1]

  dim3 g(HN / HT, DN, BN);                     // one block per (h-tile, d, b)
  FluenceVolumeLayer_19524921327963_kernel<<<g, WN, 0, stream>>>(fl, grid, pc, out);
}